// SAGELayer_3212635537936
// MI455X (gfx1250) — compile-verified
//
#include <hip/hip_runtime.h>

#define N_NODES 100000
#define N_EDGES 3200000
#define D 128
#define N_TILES ((N_NODES + 15) / 16)   // 6250, exact (100000 % 16 == 0)

typedef __attribute__((ext_vector_type(2))) float v2f;
typedef __attribute__((ext_vector_type(8))) float v8f;

// ---------------------------------------------------------------------------
// Kernel 1: zero agg[N*D] and cnt[N] in workspace (float4 stores).
// ---------------------------------------------------------------------------
__global__ void zero_ws_kernel(float4* __restrict__ ws, int n4) {
    int i = blockIdx.x * blockDim.x + threadIdx.x;
    if (i < n4) ws[i] = make_float4(0.f, 0.f, 0.f, 0.f);
}

// ---------------------------------------------------------------------------
// Kernel 2: edge scatter. One wave32 per edge; lane L covers dims [4L,4L+3]
// via a float4 gather from x[src] and 4 global f32 atomics into agg[dst].
// Lane 0 bumps the per-destination count. agg (51 MB) and x (51 MB) both fit
// in the 192 MB L2, so this runs at L2 atomic throughput, not HBM.
// ---------------------------------------------------------------------------
__global__ void sage_scatter_kernel(const float* __restrict__ x,
                                    const int* __restrict__ edges,
                                    float* __restrict__ agg,
                                    float* __restrict__ cnt) {
    int e = blockIdx.x * 8 + (threadIdx.x >> 5);   // 8 waves per 256-thread block
    if (e >= N_EDGES) return;
    int lane = threadIdx.x & 31;
    int src = edges[e];
    int dst = edges[N_EDGES + e];

    const float4* xv = (const float4*)(x + (size_t)src * D);
    float4 v = xv[lane];

    float* a = agg + (size_t)dst * D + lane * 4;
    atomicAdd(a + 0, v.x);
    atomicAdd(a + 1, v.y);
    atomicAdd(a + 2, v.z);
    atomicAdd(a + 3, v.w);
    if (lane == 0) atomicAdd(cnt + dst, 1.0f);
}

// ---------------------------------------------------------------------------
// Kernel 3: fused  out = relu( (agg/cnt) @ Wl^T + bl + x @ Wr^T )
// One wave per 16-row tile. A operands (mean tile and x tile) are kept in
// registers in the native 16x4xF32 WMMA A-layout (v2f per K-chunk per lane);
// B chunks are v2f loads from W rows (B[k,n] = W[n,k]). 64 WMMAs per N-tile.
// ---------------------------------------------------------------------------
__global__ void __launch_bounds__(256)
sage_gemm_kernel(const float* __restrict__ x,
                 const float* __restrict__ Wl,
                 const float* __restrict__ bl,
                 const float* __restrict__ Wr,
                 const float* __restrict__ agg,
                 const float* __restrict__ cnt,
                 float* __restrict__ out) {
    const int wave = threadIdx.x >> 5;            // uniform within a wave
    const int tile = blockIdx.x * 8 + wave;
    if (tile >= N_TILES) return;                  // wave-uniform: EXEC stays all-1s

    const int lane = threadIdx.x & 31;
    const int m    = lane & 15;                   // A-matrix row within tile
    const int koff = (lane >> 4) << 1;            // K offset 0 or 2
    const int row  = tile * 16 + m;               // global node row (always < N)

    const float c   = cnt[row];
    const float inv = 1.0f / fmaxf(c, 1.0f);

    // Preload A operands for both GEMMs (32 K-chunks of 4, 2 f32 each).
    v2f aL[32], aR[32];
    const float* aggRow = agg + (size_t)row * D + koff;
    const float* xRow   = x   + (size_t)row * D + koff;
#pragma unroll
    for (int kc = 0; kc < 32; ++kc) {
        v2f t = *(const v2f*)(aggRow + kc * 4);
        t.x *= inv; t.y *= inv;
        aL[kc] = t;                                // mean tile chunk
        aR[kc] = *(const v2f*)(xRow + kc * 4);     // x tile chunk
    }

    const int nl = lane & 15;                      // B/N lane index
#pragma unroll 1
    for (int nt = 0; nt < 8; ++nt) {               // 8 N-tiles of 16 cols
        const int n = nt * 16 + nl;
        const float* wl = Wl + (size_t)n * D + koff;
        const float* wr = Wr + (size_t)n * D + koff;

        v8f acc = {0.f, 0.f, 0.f, 0.f, 0.f, 0.f, 0.f, 0.f};
#pragma unroll
        for (int kc = 0; kc < 32; ++kc) {
            v2f bL = *(const v2f*)(wl + kc * 4);
            acc = __builtin_amdgcn_wmma_f32_16x16x4_f32(
                false, aL[kc], false, bL, (short)0, acc, false, false);
            v2f bR = *(const v2f*)(wr + kc * 4);
            acc = __builtin_amdgcn_wmma_f32_16x16x4_f32(
                false, aR[kc], false, bR, (short)0, acc, false, false);
        }

        // C/D layout: VGPR r -> M = r + 8*(lane>=16), N = lane&15.
        const float bias = bl[n];
        const int hi = (lane >> 4) * 8;
#pragma unroll
        for (int r = 0; r < 8; ++r) {
            float v = acc[r] + bias;
            v = fmaxf(v, 0.0f);
            out[(size_t)(tile * 16 + hi + r) * D + n] = v;
        }
    }
}

// ---------------------------------------------------------------------------
extern "C" void kernel_launch(void* const* d_in, const int* in_sizes, int n_in,
                              void* d_out, int out_size, void* d_ws, size_t ws_size,
                              hipStream_t stream) {
    const float* x     = (const float*)d_in[0];
    const int*   edges = (const int*)d_in[1];     // [2, N_EDGES], int32
    const float* Wl    = (const float*)d_in[2];
    const float* bl    = (const float*)d_in[3];
    const float* Wr    = (const float*)d_in[4];

    float* agg = (float*)d_ws;                    // N_NODES * D floats
    float* cnt = agg + (size_t)N_NODES * D;       // N_NODES floats
    float* out = (float*)d_out;

    // 1) zero workspace (12.9 M floats -> 3.225 M float4)
    const int n4 = (N_NODES * D + N_NODES) / 4;
    zero_ws_kernel<<<(n4 + 255) / 256, 256, 0, stream>>>((float4*)d_ws, n4);

    // 2) edge scatter: 8 edges (waves) per block
    sage_scatter_kernel<<<(N_EDGES + 7) / 8, 256, 0, stream>>>(x, edges, agg, cnt);

    // 3) fused mean + dual WMMA GEMM + bias + relu: 8 row-tiles per block
    sage_gemm_kernel<<<(N_TILES + 7) / 8, 256, 0, stream>>>(x, Wl, bl, Wr, agg, cnt, out);
}